// PMRLoss_9732395892833
// MI455X (gfx1250) — compile-verified
//
#include <hip/hip_runtime.h>

typedef float v2f __attribute__((ext_vector_type(2)));
typedef float v8f __attribute__((ext_vector_type(8)));

constexpr int N_ = 8192;   // batch
constexpr int C_ = 32000;  // classes
constexpr int P_ = 1024;   // prototypes
constexpr int D_ = 256;    // feature dim

// ---------- per-row sum of squares over a 256-wide row ----------
__global__ void rowsq256_kernel(const float* __restrict__ x, float* __restrict__ out) {
    const int row = blockIdx.x;
    const int t   = threadIdx.x;
    float v = x[(size_t)row * D_ + t];
    v *= v;
    __shared__ float sm[256];
    sm[t] = v;
    __syncthreads();
    for (int off = 128; off > 0; off >>= 1) {
        if (t < off) sm[t] += sm[t + off];
        __syncthreads();
    }
    if (t == 0) out[row] = sm[0];
}

// ---------- cross-entropy: one block per row, single-pass online LSE ----------
// Online update amortized per float4: only one branch + (rare) rescale per 4
// elements; common path is exactly one v_exp_f32 per element so the kernel
// stays HBM-bound (1.05 GB @ 23.3 TB/s), not transcendental-bound.
__global__ void ce_rows_kernel(const float* __restrict__ logits,
                               const int* __restrict__ targets,
                               float* __restrict__ ws_ce) {
    const int row = blockIdx.x;
    const int t   = threadIdx.x;
    const float4* lp = (const float4*)(logits + (size_t)row * C_);

    float m = -INFINITY, s = 0.f;
    for (int i = t; i < C_ / 4; i += 256) {
        float4 v = lp[i];
        float m4 = fmaxf(fmaxf(v.x, v.y), fmaxf(v.z, v.w));
        if (m4 > m) {              // rare after warm-up; s==0 case: 0 * exp(-inf) == 0
            s *= __expf(m - m4);
            m = m4;
        }
        s += (__expf(v.x - m) + __expf(v.y - m)) + (__expf(v.z - m) + __expf(v.w - m));
    }

    __shared__ float smM[256], smS[256];
    smM[t] = m; smS[t] = s;
    __syncthreads();
    for (int off = 128; off > 0; off >>= 1) {
        if (t < off) {
            float ma = smM[t],       sa = smS[t];
            float mb = smM[t + off], sb = smS[t + off];
            float M = fmaxf(ma, mb);
            smM[t] = M;
            smS[t] = sa * __expf(ma - M) + sb * __expf(mb - M);
        }
        __syncthreads();
    }
    if (t == 0) {
        int   tgt = targets[row];
        float xt  = logits[(size_t)row * C_ + tgt];
        ws_ce[row] = -(xt - smM[0] - __logf(smS[0]));
    }
}

// ---------- prototype loss: f32 WMMA GEMM tiles + fused exp-sum epilogue ----------
// Grid: N_/16 row-tile blocks, 256 threads (8 waves). Wave w covers column
// tiles [w*8, w*8+8) of 16 prototypes each; K loop in steps of 4 using
// V_WMMA_F32_16X16X4_F32 (full f32 precision, matches reference).
__global__ void proto_rows_kernel(const float* __restrict__ features,
                                  const float* __restrict__ prototypes,
                                  const float* __restrict__ f2,
                                  const float* __restrict__ p2,
                                  float* __restrict__ ws_proto) {
    const int rt   = blockIdx.x;       // row tile 0..511
    const int t    = threadIdx.x;      // 0..255
    const int w    = t >> 5;           // wave 0..7
    const int lane = t & 31;
    const int half = lane >> 4;        // K-half selector (A/B layout)
    const int lr   = lane & 15;        // row (A) / col (B) within tile

    const float* arow = features + (size_t)(rt * 16 + lr) * D_;

    float acc[8];
#pragma unroll
    for (int i = 0; i < 8; ++i) acc[i] = 0.f;

    for (int j = 0; j < 8; ++j) {
        const int ct = w * 8 + j;             // column tile 0..63
        const int n  = ct * 16 + lr;          // prototype index this lane loads
        const float* brow = prototypes + (size_t)n * D_;
        const float  p2n  = p2[n];

        v8f c = {};
        for (int k0 = 0; k0 < D_; k0 += 4) {
            const int kk = k0 + half * 2;     // lanes 0-15: K={k0,k0+1}; 16-31: K={k0+2,k0+3}
            v2f a = *(const v2f*)(arow + kk);
            v2f b = *(const v2f*)(brow + kk);
            c = __builtin_amdgcn_wmma_f32_16x16x4_f32(
                    false, a, false, b, (short)0, c, false, false);
        }
        // c[i] = dot(features[rt*16 + half*8 + i], prototypes[ct*16 + lr])
        // exp(-d2) = exp(-f2)*exp(2c - p2); f2 factor applied after reduction.
#pragma unroll
        for (int i = 0; i < 8; ++i)
            acc[i] += __expf(2.f * c[i] - p2n);
    }

    // Deterministic butterfly sum over the 16 lanes of each half (over N).
#pragma unroll
    for (int i = 0; i < 8; ++i) {
        float v = acc[i];
        v += __shfl_xor(v, 1, 32);
        v += __shfl_xor(v, 2, 32);
        v += __shfl_xor(v, 4, 32);
        v += __shfl_xor(v, 8, 32);
        acc[i] = v;
    }

    __shared__ float sm[8][16];
    if (lr == 0) {
#pragma unroll
        for (int i = 0; i < 8; ++i) sm[w][half * 8 + i] = acc[i];
    }
    __syncthreads();

    if (t < 16) {   // one thread per output row of this tile; fixed-order sum over waves
        float S = 0.f;
#pragma unroll
        for (int wv = 0; wv < 8; ++wv) S += sm[wv][t];
        const int row = rt * 16 + t;
        // -log( exp(-f2) * S ) = f2 - log(S)
        ws_proto[row] = f2[row] - __logf(S);
    }
}

// ---------- final deterministic reduction to (total, ce, proto) ----------
__global__ void final_reduce_kernel(const float* __restrict__ ws_ce,
                                    const float* __restrict__ ws_proto,
                                    float* __restrict__ out) {
    const int t = threadIdx.x;
    float sc = 0.f, sp = 0.f;
    for (int i = t; i < N_; i += 256) { sc += ws_ce[i]; sp += ws_proto[i]; }
    __shared__ float smc[256], smp[256];
    smc[t] = sc; smp[t] = sp;
    __syncthreads();
    for (int off = 128; off > 0; off >>= 1) {
        if (t < off) { smc[t] += smc[t + off]; smp[t] += smp[t + off]; }
        __syncthreads();
    }
    if (t == 0) {
        float ce    = smc[0] / (float)N_;
        float proto = smp[0] / (float)N_;
        out[0] = ce + 0.001f * proto;  // total = LAMBDA_CE*ce + LAMBDA_PROTO*proto
        out[1] = ce;
        out[2] = proto;
    }
}

extern "C" void kernel_launch(void* const* d_in, const int* in_sizes, int n_in,
                              void* d_out, int out_size, void* d_ws, size_t ws_size,
                              hipStream_t stream) {
    const float* logits     = (const float*)d_in[0];
    const float* prototypes = (const float*)d_in[1];
    const float* features   = (const float*)d_in[2];
    const int*   targets    = (const int*)d_in[3];
    float*       out        = (float*)d_out;

    float* ws       = (float*)d_ws;
    float* f2       = ws;                 // [N_]
    float* p2       = ws + N_;            // [P_]
    float* ws_ce    = ws + N_ + P_;       // [N_]
    float* ws_proto = ws + 2 * N_ + P_;   // [N_]

    rowsq256_kernel<<<N_, 256, 0, stream>>>(features, f2);
    rowsq256_kernel<<<P_, 256, 0, stream>>>(prototypes, p2);
    ce_rows_kernel<<<N_, 256, 0, stream>>>(logits, targets, ws_ce);
    proto_rows_kernel<<<N_ / 16, 256, 0, stream>>>(features, prototypes, f2, p2, ws_proto);
    final_reduce_kernel<<<1, 256, 0, stream>>>(ws_ce, ws_proto, out);
}